// SparseGraphAttention_6528350289922
// MI455X (gfx1250) — compile-verified
//
#include <hip/hip_runtime.h>
#include <hip/hip_bf16.h>

#define D_MODEL   128
#define N_HEADS   8
#define HEAD_DIM  16
#define SCALE     0.25f   // HEAD_DIM^-0.5

typedef __attribute__((ext_vector_type(2))) float v2f;
typedef __attribute__((ext_vector_type(8))) float v8f;
typedef __attribute__((ext_vector_type(2))) int   v2i;

typedef __attribute__((address_space(1))) v2i g_v2i_t;   // global  (device) v2i
typedef __attribute__((address_space(3))) v2i l_v2i_t;   // LDS (shared) v2i

// ---------------------------------------------------------------------------
// Zero-init kernel (workspace is poisoned by harness; acc/att_sum need zeros)
// ---------------------------------------------------------------------------
__global__ void sga_zero_kernel(float* __restrict__ p, int n) {
    int i = blockIdx.x * blockDim.x + threadIdx.x;
    if (i < n) p[i] = 0.0f;
}

// ---------------------------------------------------------------------------
// NT GEMM with FP32 WMMA: C[M,N] = A[M,K] @ B[N,K]^T, K = 128 fixed.
//
// Block = 256 threads = 8 waves. blockIdx.x -> 128-row M block (8 x 16-row
// wave tiles), blockIdx.y -> 64-col N strip. The 64x128 B strip (32KB) is
// staged once into LDS in WMMA-fragment order and shared by all 8 waves:
//   bs[((t*(K/4) + ks)*32 + lane)*2 + {0,1}]
//     = B[col0 + t*16 + (lane&15)][ks*4 + (lane>>4)*2 + {0,1}]
// so the compute-loop B read is a conflict-free contiguous ds_load_b64.
// Fill uses async global->LDS (ASYNCcnt path) when the builtin exists.
//
// A fragment (16x4 f32, 2 VGPRs): lane L holds A[row0+(L&15)][k+(L>>4)*2+{0,1}]
// and is double-buffered in registers: the global load for step ks+1 issues
// before the 4 WMMAs consuming step ks.
// D (16x16 f32, 8 VGPRs): lane L, vgpr r -> M = r + 8*(L>>4), N = (L&15)
// ---------------------------------------------------------------------------
template<int TN, int K>
__global__ void __launch_bounds__(256) sga_wmma_gemm_nt(const float* __restrict__ A,
                                                        const float* __restrict__ B,
                                                        float* __restrict__ C,
                                                        int M, int N) {
    constexpr int KS = K / 4;              // K-steps (32)
    constexpr int NF = TN * KS;            // fragments per strip (128)
    __shared__ float bs[NF * 32 * 2];      // 32KB for TN=4, K=128

    const int tid   = threadIdx.x;
    const int lane  = tid & 31;
    const int wave  = tid >> 5;            // 0..7
    const int r     = lane & 15;           // row (A) / col (B) within tile
    const int kh2   = (lane >> 4) * 2;     // K sub-offset: 0 or 2

    const int mTiles = M >> 4;
    const int mTile  = blockIdx.x * 8 + wave;
    const int col0   = blockIdx.y * 16 * TN;

    // ---- cooperative B-strip fill: 8 fragments per pass, 16 passes --------
    for (int f = wave; f < NF; f += 8) {
        const int t  = f / KS;
        const int ks = f % KS;
        const float* src = B + (size_t)(col0 + t * 16 + r) * K + ks * 4 + kh2;
        float*       dst = &bs[(size_t)(f * 32 + lane) * 2];
#if __has_builtin(__builtin_amdgcn_global_load_async_to_lds_b64)
        __builtin_amdgcn_global_load_async_to_lds_b64((g_v2i_t*)src, (l_v2i_t*)dst,
                                                      /*offset=*/0, /*cpol=*/0);
#else
        *(v2f*)dst = *(const v2f*)src;
#endif
    }
#if __has_builtin(__builtin_amdgcn_global_load_async_to_lds_b64)
#if __has_builtin(__builtin_amdgcn_s_wait_asynccnt)
    __builtin_amdgcn_s_wait_asynccnt(0);
#else
    asm volatile("s_wait_asynccnt 0x0" ::: "memory");
#endif
#endif
    __syncthreads();

    if (mTile < mTiles) {                  // wave-uniform: EXEC stays full
        const int row0 = mTile * 16;
        const float* aRow = A + (size_t)(row0 + r) * K + kh2;

        v8f acc[TN] = {};
        v2f a_cur = *(const v2f*)(aRow);
        v2f a_next = a_cur;

#pragma unroll 4
        for (int ks = 0; ks < KS; ++ks) {
            if (ks + 1 < KS)               // prefetch next A fragment (overlaps WMMAs)
                a_next = *(const v2f*)(aRow + (ks + 1) * 4);
#pragma unroll
            for (int t = 0; t < TN; ++t) {
                v2f b = *(const v2f*)&bs[(size_t)((t * KS + ks) * 32 + lane) * 2];
                acc[t] = __builtin_amdgcn_wmma_f32_16x16x4_f32(
                    /*neg_a=*/false, a_cur, /*neg_b=*/false, b,
                    /*c_mod=*/(short)0, acc[t], /*reuse_a=*/false, /*reuse_b=*/false);
            }
            a_cur = a_next;
        }

        const int mbase = row0 + (lane >> 4) * 8;
#pragma unroll
        for (int t = 0; t < TN; ++t) {
            float* cp = C + (size_t)mbase * N + col0 + t * 16 + (lane & 15);
#pragma unroll
            for (int rr = 0; rr < 8; ++rr) {
                cp[(size_t)rr * N] = acc[t][rr];
            }
        }
    }
}

// ---------------------------------------------------------------------------
// Edge attention: one thread per (edge, head).
// qkv layout per node: [q(128) | k(128) | v(128)], head h occupies 16 floats.
// Gathers hit L2 (76.8MB qkv fits in 192MB L2); segment-sum via L2 atomics.
// ---------------------------------------------------------------------------
__global__ void sga_edge_kernel(const float* __restrict__ qkv,
                                const int* __restrict__ edge_src,
                                const int* __restrict__ edge_tgt,
                                float* __restrict__ acc,       // [N, 128]
                                float* __restrict__ att_sum,   // [N, 8]
                                int E) {
    int tid = blockIdx.x * blockDim.x + threadIdx.x;
    if (tid >= E * N_HEADS) return;
    const int e = tid >> 3;
    const int h = tid & 7;

    const int src = edge_src[e];
    const int tgt = edge_tgt[e];

    const float* qp = qkv + (size_t)src * 384 + h * HEAD_DIM;          // q
    const float* kp = qkv + (size_t)tgt * 384 + 128 + h * HEAD_DIM;    // k
    const float* vp = qkv + (size_t)tgt * 384 + 256 + h * HEAD_DIM;    // v

    float4 q0 = *(const float4*)(qp + 0),  q1 = *(const float4*)(qp + 4);
    float4 q2 = *(const float4*)(qp + 8),  q3 = *(const float4*)(qp + 12);
    float4 k0 = *(const float4*)(kp + 0),  k1 = *(const float4*)(kp + 4);
    float4 k2 = *(const float4*)(kp + 8),  k3 = *(const float4*)(kp + 12);

    float dot = q0.x*k0.x + q0.y*k0.y + q0.z*k0.z + q0.w*k0.w
              + q1.x*k1.x + q1.y*k1.y + q1.z*k1.z + q1.w*k1.w
              + q2.x*k2.x + q2.y*k2.y + q2.z*k2.z + q2.w*k2.w
              + q3.x*k3.x + q3.y*k3.y + q3.z*k3.z + q3.w*k3.w;

    float s = fminf(fmaxf(dot * SCALE, -10.0f), 10.0f);
    float w = __expf(s);

    atomicAdd(att_sum + (size_t)src * N_HEADS + h, w);

    float4 v0 = *(const float4*)(vp + 0),  v1 = *(const float4*)(vp + 4);
    float4 v2 = *(const float4*)(vp + 8),  v3 = *(const float4*)(vp + 12);

    float* ap = acc + (size_t)src * D_MODEL + h * HEAD_DIM;
    atomicAdd(ap + 0,  w * v0.x); atomicAdd(ap + 1,  w * v0.y);
    atomicAdd(ap + 2,  w * v0.z); atomicAdd(ap + 3,  w * v0.w);
    atomicAdd(ap + 4,  w * v1.x); atomicAdd(ap + 5,  w * v1.y);
    atomicAdd(ap + 6,  w * v1.z); atomicAdd(ap + 7,  w * v1.w);
    atomicAdd(ap + 8,  w * v2.x); atomicAdd(ap + 9,  w * v2.y);
    atomicAdd(ap + 10, w * v2.z); atomicAdd(ap + 11, w * v2.w);
    atomicAdd(ap + 12, w * v3.x); atomicAdd(ap + 13, w * v3.y);
    atomicAdd(ap + 14, w * v3.z); atomicAdd(ap + 15, w * v3.w);
}

// ---------------------------------------------------------------------------
// In-place normalize: acc[i, h*16+d] /= (att_sum[i, h] + 1e-8)
// ---------------------------------------------------------------------------
__global__ void sga_normalize_kernel(float* __restrict__ acc,
                                     const float* __restrict__ att_sum,
                                     int total /* N*128 */) {
    int i = blockIdx.x * blockDim.x + threadIdx.x;
    if (i >= total) return;
    int node = i >> 7;           // /128
    int h    = (i >> 4) & 7;     // head
    acc[i] = acc[i] / (att_sum[node * N_HEADS + h] + 1e-8f);
}

// ---------------------------------------------------------------------------
// Launch
// ---------------------------------------------------------------------------
extern "C" void kernel_launch(void* const* d_in, const int* in_sizes, int n_in,
                              void* d_out, int out_size, void* d_ws, size_t ws_size,
                              hipStream_t stream) {
    const float* x     = (const float*)d_in[0];   // node_embeds [N,128]
    const int*   eidx  = (const int*)  d_in[1];   // edge_index  [2,E]
    const float* wqkv  = (const float*)d_in[2];   // W_qkv [384,128]
    const float* wout  = (const float*)d_in[3];   // W_out [128,128]
    float*       out   = (float*)d_out;           // [N,128]

    const int N = in_sizes[0] / D_MODEL;          // 50000
    const int E = in_sizes[1] / 2;                // 800000

    // Workspace: qkv [N,384] | acc [N,128] | att_sum [N,8]
    float* qkv  = (float*)d_ws;
    float* acc  = qkv + (size_t)N * 384;
    float* asum = acc + (size_t)N * D_MODEL;

    const int mBlocks = (N / 16 + 7) / 8;         // 8 waves (16-row tiles) per block

    // 1) zero acc + att_sum (contiguous: N*136 floats)
    {
        int n = N * (D_MODEL + N_HEADS);
        sga_zero_kernel<<<(n + 255) / 256, 256, 0, stream>>>(acc, n);
    }

    // 2) QKV projection: qkv = X @ W_qkv^T   (M=N, N=384, K=128)
    {
        dim3 grid(mBlocks, 384 / 64);
        sga_wmma_gemm_nt<4, D_MODEL><<<grid, 256, 0, stream>>>(x, wqkv, qkv, N, 384);
    }

    // 3) edge attention (gather + exp + L2 atomic segment-sum)
    {
        int n = E * N_HEADS;
        sga_edge_kernel<<<(n + 255) / 256, 256, 0, stream>>>(
            qkv, eidx, eidx + E, acc, asum, E);
    }

    // 4) normalize in place
    {
        int n = N * D_MODEL;
        sga_normalize_kernel<<<(n + 255) / 256, 256, 0, stream>>>(acc, asum, n);
    }

    // 5) output projection: out = acc @ W_out^T  (M=N, N=128, K=128)
    {
        dim3 grid(mBlocks, D_MODEL / 64);
        sga_wmma_gemm_nt<4, D_MODEL><<<grid, 256, 0, stream>>>(acc, wout, out, N, D_MODEL);
    }
}